// SSLPretrainModel_36026185679272
// MI455X (gfx1250) — compile-verified
//
#include <hip/hip_runtime.h>

typedef float v2f __attribute__((ext_vector_type(2)));
typedef float v4f __attribute__((ext_vector_type(4)));
typedef float v8f __attribute__((ext_vector_type(8)));

#define TB 256
#define WPB 8  // waves per block (wave32)

// ---------------------------------------------------------------------------
// Wave-strip f32 WMMA core, K stepped by 8 (2 WMMAs per step, b128 loads).
// K-permutation trick: WMMA instr j consumes matrix-K {8j,8j+1,8j+4,8j+5},
// instr j' consumes {8j+2,8j+3,8j+6,8j+7}; lanes<16 carry the first pair,
// lanes>=16 the second, so one float4 at A[m][8j+4*half] (and Bt[n][same])
// feeds both instructions. Bt is transposed+zero-padded to Kpad (mult of 8),
// so K-tail garbage in A is annihilated by zero B columns.
// ---------------------------------------------------------------------------
template <int NT, bool ROWG, bool COLG, class FF, class FT>
__device__ __forceinline__ void wmma_core(
    FF ldA_fast, FT ldA_tail,
    const float* __restrict__ Bt, int ldbt, int N,
    int Kfast, int Kpad,
    int m0, int M,
    const float* __restrict__ addmat, int ldadd,
    const float* __restrict__ bias,
    float* __restrict__ out, int ldc, float* __restrict__ out_pre, bool relu)
{
    const int lane = threadIdx.x & 31;
    const int half = lane >> 4;
    const int mn   = lane & 15;
    const int hb   = half << 2;   // 0 or 4

    int bOff[NT];
#pragma unroll
    for (int t = 0; t < NT; ++t) {
        int n = t * 16 + mn;
        if (n > N - 1) n = N - 1;          // clamp (discarded by store guard)
        bOff[t] = n * ldbt;
    }

    v8f acc[NT];
#pragma unroll
    for (int t = 0; t < NT; ++t)
#pragma unroll
        for (int r = 0; r < 8; ++r) acc[t][r] = 0.0f;

    auto step = [&](int kb, v4f a) {
        v2f a01; a01.x = a.x; a01.y = a.y;
        v2f a23; a23.x = a.z; a23.y = a.w;
#pragma unroll
        for (int t = 0; t < NT; ++t) {
            v4f b = *(const v4f*)(Bt + bOff[t] + kb);
            v2f b01; b01.x = b.x; b01.y = b.y;
            v2f b23; b23.x = b.z; b23.y = b.w;
            acc[t] = __builtin_amdgcn_wmma_f32_16x16x4_f32(
                false, a01, false, b01, (short)0, acc[t], false, false);
            acc[t] = __builtin_amdgcn_wmma_f32_16x16x4_f32(
                false, a23, false, b23, (short)0, acc[t], false, false);
        }
    };

    for (int k = 0; k < Kfast; k += 8) { const int kb = k + hb; step(kb, ldA_fast(kb)); }
    for (int k = Kfast; k < Kpad; k += 8) { const int kb = k + hb; step(kb, ldA_tail(kb)); }

#pragma unroll
    for (int t = 0; t < NT; ++t) {
        const int col = t * 16 + mn;
        const int colc = (COLG && col > N - 1) ? N - 1 : col;
        const float bv = bias ? bias[colc] : 0.0f;
#pragma unroll
        for (int r = 0; r < 8; ++r) {
            const int row = m0 + (half << 3) + r;
            const bool ok = (!ROWG || row < M) && (!COLG || col < N);
            if (ok) {
                float v = acc[t][r] + bv;
                if (addmat) v += addmat[(size_t)row * ldadd + col];
                if (out_pre) out_pre[(size_t)row * ldc + col] = v;
                out[(size_t)row * ldc + col] = relu ? fmaxf(v, 0.0f) : v;
            }
        }
    }
}

// -------------------- generic row-major-A GEMM --------------------
template <int NT, bool ALN, bool ROWG, bool COLG>
__global__ __launch_bounds__(TB) void k_gemm_rowA(
    const float* __restrict__ A, int lda,
    const float* __restrict__ Bt, int ldbt,
    int M, int N, int K, int Kpad,
    const float* __restrict__ addmat, int ldadd,
    const float* __restrict__ bias,
    float* __restrict__ out, float* __restrict__ out_pre, int ldc, int relu)
{
    const int w  = blockIdx.x * WPB + (threadIdx.x >> 5);
    const int m0 = w << 4;
    int m = m0 + (threadIdx.x & 15);
    if (m > M - 1) m = M - 1;                    // clamp: loads stay in-bounds
    const float* Ap = A + (size_t)m * lda;
    const int Kc = K - 1;

    auto ld_fast = [&](int kb) -> v4f {
        if (ALN) {
            return *(const v4f*)(Ap + kb);
        } else {
            v4f a; __builtin_memcpy(&a, Ap + kb, sizeof(v4f));
            return a;
        }
    };
    auto ld_tail = [&](int kb) -> v4f {         // clamped; garbage * 0-pad B
        v4f a;
        a.x = Ap[(kb     > Kc) ? Kc : kb];
        a.y = Ap[(kb + 1 > Kc) ? Kc : kb + 1];
        a.z = Ap[(kb + 2 > Kc) ? Kc : kb + 2];
        a.w = Ap[(kb + 3 > Kc) ? Kc : kb + 3];
        return a;
    };
    wmma_core<NT, ROWG, COLG>(ld_fast, ld_tail, Bt, ldbt, N, K & ~7, Kpad,
                              m0, M, addmat, ldadd, bias, out, ldc, out_pre, relu != 0);
}

// msg_new = relu(inp + (a_msg[src[e]] - msg[rev[e]]) @ W_h)   (H = 256)
__global__ __launch_bounds__(TB) void k_gemm_dmpnn(
    const float* __restrict__ a_msg, const float* __restrict__ msg,
    const int* __restrict__ src, const int* __restrict__ rev,
    const float* __restrict__ Wt_h, const float* __restrict__ inp,
    int E, float* __restrict__ out)
{
    const int w  = blockIdx.x * WPB + (threadIdx.x >> 5);
    const int m0 = w << 4;
    int e = m0 + (threadIdx.x & 15);
    if (e > E - 1) e = E - 1;
    const float* pa = a_msg + (size_t)src[e] * 256;
    const float* pm = msg   + (size_t)rev[e] * 256;
    auto ld = [&](int kb) -> v4f {
        return *(const v4f*)(pa + kb) - *(const v4f*)(pm + kb);
    };
    wmma_core<16, false, false>(ld, ld, Wt_h, 256, 256, 256, 256,
                                m0, E, inp, 256, nullptr, out, 256, nullptr, true);
}

// edge_pred = (0.5*(ah[src[2i]] + ah[dst[2i]])) @ edge_W + edge_b, N = 14
__global__ __launch_bounds__(TB) void k_gemm_edge(
    const float* __restrict__ ah,
    const int* __restrict__ src, const int* __restrict__ dst,
    const float* __restrict__ Wt_e, const float* __restrict__ edge_b,
    int Eh, float* __restrict__ out)
{
    const int w  = blockIdx.x * WPB + (threadIdx.x >> 5);
    const int m0 = w << 4;
    int i = m0 + (threadIdx.x & 15);
    if (i > Eh - 1) i = Eh - 1;
    const int e = i << 1;
    const float* ps = ah + (size_t)src[e] * 256;
    const float* pd = ah + (size_t)dst[e] * 256;
    auto ld = [&](int kb) -> v4f {
        return (*(const v4f*)(ps + kb) + *(const v4f*)(pd + kb)) * 0.5f;
    };
    wmma_core<1, false, true>(ld, ld, Wt_e, 256, 14, 256, 256,
                              m0, Eh, nullptr, 0, edge_b, out, 14, nullptr, false);
}

// -------------------- weight transpose + zero-pad: W[K,N] -> Wt[N,Kpad] ----
__global__ __launch_bounds__(TB) void k_transpose_pad(
    const float* __restrict__ W, int K, int N, int Kpad, float* __restrict__ Wt)
{
    const int i = blockIdx.x * TB + threadIdx.x;
    if (i >= N * Kpad) return;
    const int n = i / Kpad, k = i - n * Kpad;
    Wt[i] = (k < K) ? W[(size_t)k * N + n] : 0.0f;
}

// -------------------- segment-sum scatter (rows of 256 floats) -------------
__global__ __launch_bounds__(TB) void k_scatter_rows(
    const float* __restrict__ rows, const int* __restrict__ seg,
    float* __restrict__ out, int nrows)
{
    const int e = blockIdx.x * 4 + (threadIdx.x >> 6);   // 64 threads per row
    if (e >= nrows) return;
    const int c = (threadIdx.x & 63) << 2;
    const int s = seg[e];
    const float4 v = *(const float4*)(rows + (size_t)e * 256 + c);
    float* o = out + (size_t)s * 256 + c;
    atomicAdd(o + 0, v.x); atomicAdd(o + 1, v.y);
    atomicAdd(o + 2, v.z); atomicAdd(o + 3, v.w);
}

// -------------------- graph head final matvec -------------------------------
__global__ __launch_bounds__(TB) void k_graph_matvec(
    const float* __restrict__ h, const float* __restrict__ w2,
    const float* __restrict__ b2, float* __restrict__ out, int M)
{
    const int m = blockIdx.x * blockDim.x + threadIdx.x;
    if (m >= M) return;
    const float* p = h + (size_t)m * 256;
    float s = 0.0f;
    for (int k = 0; k < 256; ++k) s = fmaf(p[k], w2[k], s);
    out[m] = s + b2[0];
}

// ---------------------------------------------------------------------------
extern "C" void kernel_launch(void* const* d_in, const int* in_sizes, int n_in,
                              void* d_out, int out_size, void* d_ws, size_t ws_size,
                              hipStream_t stream) {
    const float* f_atoms = (const float*)d_in[0];
    const float* f_bonds = (const float*)d_in[1];
    const float* W_i     = (const float*)d_in[2];
    const float* W_h     = (const float*)d_in[3];
    const float* W_o     = (const float*)d_in[4];
    const float* b_o     = (const float*)d_in[5];
    const float* node_W  = (const float*)d_in[6];
    const float* node_b  = (const float*)d_in[7];
    const float* edge_W  = (const float*)d_in[8];
    const float* edge_b  = (const float*)d_in[9];
    const float* g_W1    = (const float*)d_in[10];
    const float* g_b1    = (const float*)d_in[11];
    const float* g_W2    = (const float*)d_in[12];
    const float* g_b2    = (const float*)d_in[13];
    const int* edge_index = (const int*)d_in[14];
    const int* rev        = (const int*)d_in[15];
    const int* atom_mol   = (const int*)d_in[16];

    const int ATOMF = 133, BONDF = 14, BFD = 147, H = 256;
    const int Na = in_sizes[0] / ATOMF;                       // 400000
    const int E  = in_sizes[1] / BFD;                         // 800000
    const int M  = out_size - Na * ATOMF - (E / 2) * BONDF;   // 13000
    const int* src = edge_index;
    const int* dst = edge_index + E;
    const int KP_I = 152;   // 147 -> pad8
    const int KP_T = 136;   // 133 -> pad8

    // workspace layout (fp32)
    float* ws = (float*)d_ws;
    size_t off = 0;
    float* inp    = ws + off; off += (size_t)E  * H;
    float* msgA   = ws + off; off += (size_t)E  * H;
    float* msgB   = ws + off; off += (size_t)E  * H;
    float* a_msg  = ws + off; off += (size_t)Na * H;
    float* atom_h = ws + off; off += (size_t)Na * H;
    float* g_emb  = ws + off; off += (size_t)M  * H;
    float* g_hid  = ws + off; off += (size_t)M  * H;
    float* Wt_i   = ws + off; off += (size_t)H * KP_I;   // [256][152]
    float* Wt_h   = ws + off; off += (size_t)H * H;      // [256][256]
    float* Wt_ot  = ws + off; off += (size_t)H * KP_T;   // W_o top  [256][136]
    float* Wt_ob  = ws + off; off += (size_t)H * H;      // W_o bot  [256][256]
    float* Wt_n   = ws + off; off += (size_t)ATOMF * H;  // [133][256]
    float* Wt_e   = ws + off; off += (size_t)BONDF * H;  // [14][256]
    float* Wt_g1  = ws + off; off += (size_t)H * H;      // [256][256]
    (void)off; (void)ws_size; (void)n_in;

    float* out_node  = (float*)d_out;
    float* out_edge  = out_node + (size_t)Na * ATOMF;
    float* out_graph = out_edge + (size_t)(E / 2) * BONDF;

    const dim3 tb(TB);
    const int gE  = (E / 16) / WPB;                 // 6250
    const int gNa = (Na / 16) / WPB;                // 3125
    const int gEd = ((E / 2) / 16) / WPB;           // 3125
    const int gM  = ((M + 15) / 16 + WPB - 1) / WPB;
#define TGRID(n) (((n) + TB - 1) / TB)

    // 0) transpose + zero-pad all weight matrices (tiny)
    k_transpose_pad<<<TGRID(H * KP_I), tb, 0, stream>>>(W_i, BFD, H, KP_I, Wt_i);
    k_transpose_pad<<<TGRID(H * H),    tb, 0, stream>>>(W_h, H, H, H, Wt_h);
    k_transpose_pad<<<TGRID(H * KP_T), tb, 0, stream>>>(W_o, ATOMF, H, KP_T, Wt_ot);
    k_transpose_pad<<<TGRID(H * H),    tb, 0, stream>>>(W_o + (size_t)ATOMF * H, H, H, H, Wt_ob);
    k_transpose_pad<<<TGRID(ATOMF * H), tb, 0, stream>>>(node_W, H, ATOMF, H, Wt_n);
    k_transpose_pad<<<TGRID(BONDF * H), tb, 0, stream>>>(edge_W, H, BONDF, H, Wt_e);
    k_transpose_pad<<<TGRID(H * H),    tb, 0, stream>>>(g_W1, H, H, H, Wt_g1);

    // 1) inp = f_bonds @ W_i ; msg = relu(inp)
    k_gemm_rowA<16, false, false, false><<<gE, tb, 0, stream>>>(
        f_bonds, BFD, Wt_i, KP_I, E, H, BFD, KP_I,
        nullptr, 0, nullptr, msgA, inp, H, 1);

    // 2) two message-passing rounds
    float* cur = msgA; float* nxt = msgB;
    for (int it = 0; it < 2; ++it) {
        hipMemsetAsync(a_msg, 0, (size_t)Na * H * sizeof(float), stream);
        k_scatter_rows<<<E / 4, tb, 0, stream>>>(cur, dst, a_msg, E);
        k_gemm_dmpnn<<<gE, tb, 0, stream>>>(a_msg, cur, src, rev, Wt_h, inp, E, nxt);
        float* t = cur; cur = nxt; nxt = t;
    }

    // 3) final aggregation + atom hiddens (concat GEMM split in two)
    hipMemsetAsync(a_msg, 0, (size_t)Na * H * sizeof(float), stream);
    k_scatter_rows<<<E / 4, tb, 0, stream>>>(cur, dst, a_msg, E);
    k_gemm_rowA<16, true, false, false><<<gNa, tb, 0, stream>>>(   // a_msg @ W_o[133:]
        a_msg, H, Wt_ob, H, Na, H, H, H,
        nullptr, 0, nullptr, atom_h, nullptr, H, 0);
    k_gemm_rowA<16, false, false, false><<<gNa, tb, 0, stream>>>(  // + f_atoms @ W_o[:133] + b_o, relu
        f_atoms, ATOMF, Wt_ot, KP_T, Na, H, ATOMF, KP_T,
        atom_h, H, b_o, atom_h, nullptr, H, 1);

    // 4) graph pooling
    hipMemsetAsync(g_emb, 0, (size_t)M * H * sizeof(float), stream);
    k_scatter_rows<<<(Na + 3) / 4, tb, 0, stream>>>(atom_h, atom_mol, g_emb, Na);

    // 5) node head (N=133 -> 9 col tiles, col guard)
    k_gemm_rowA<9, true, false, true><<<gNa, tb, 0, stream>>>(
        atom_h, H, Wt_n, H, Na, ATOMF, H, H,
        nullptr, 0, node_b, out_node, nullptr, ATOMF, 0);

    // 6) edge head (dedup: even bonds only)
    k_gemm_edge<<<gEd, tb, 0, stream>>>(atom_h, src, dst, Wt_e, edge_b, E / 2, out_edge);

    // 7) graph head
    k_gemm_rowA<16, true, true, false><<<gM, tb, 0, stream>>>(
        g_emb, H, Wt_g1, H, M, H, H, H,
        nullptr, 0, g_b1, g_hid, nullptr, H, 1);
    k_graph_matvec<<<(M + TB - 1) / TB, tb, 0, stream>>>(g_hid, g_W2, g_b2, out_graph, M);
#undef TGRID
}